// GCNResnet_58789512348389
// MI455X (gfx1250) — compile-verified
//
#include <hip/hip_runtime.h>

// CDNA5 / gfx1250: fp32 WMMA flash-attention for
//   v = softmax(pooled @ pooled^T) @ pooled,  pooled = f(x, A, W)  [B=8192, d=10]

typedef float v2f __attribute__((ext_vector_type(2)));
typedef float v8f __attribute__((ext_vector_type(8)));

__device__ __forceinline__ v8f wmma4(v2f a, v2f b, v8f c) {
  // D = A(16x4) * B(4x16) + C, fp32 in/out
  return __builtin_amdgcn_wmma_f32_16x16x4_f32(false, a, false, b, (short)0, c,
                                               false, false);
}

// ---------------------------------------------------------------------------
// Kernel 1: pooled[b, :] = (sum_m c_m * x[b,m,:]) @ W, padded to 16 dims.
//   dims 10..14 = 0, dim 15 = 1.0  (ones column: constant +1 logit shift for
//   QK^T, and free softmax-denominator accumulation in the PV matmul).
// Stores row-major pp[B][16] and tile-transposed ppT[B/16][16 dims][16 keys].
// ---------------------------------------------------------------------------
__global__ __launch_bounds__(256) void pooled_kernel(
    const float* __restrict__ x, const float* __restrict__ A,
    const float* __restrict__ W, float* __restrict__ pp,
    float* __restrict__ ppT, int B) {
  int b = blockIdx.x * blockDim.x + threadIdx.x;
  if (b >= B) return;
  // adj = 0.25*(off + A); pooled weight c_m = 0.125*(a[0,m] + a[1,m])
  // off[0]+off[1] = {1, 1, 2}
  float c0 = 0.125f * (1.0f + A[0] + A[3]);
  float c1 = 0.125f * (1.0f + A[1] + A[4]);
  float c2 = 0.125f * (2.0f + A[2] + A[5]);
  const float* xb = x + b * 30;
  float y[10];
#pragma unroll
  for (int f = 0; f < 10; ++f)
    y[f] = c0 * xb[f] + c1 * xb[10 + f] + c2 * xb[20 + f];

  float* row = pp + b * 16;
  float* col = ppT + (b >> 4) * 256 + (b & 15);
#pragma unroll
  for (int o = 0; o < 10; ++o) {
    float s = 0.f;
#pragma unroll
    for (int f = 0; f < 10; ++f) s += y[f] * W[f * 10 + o];
    row[o] = s;
    col[o * 16] = s;
  }
#pragma unroll
  for (int o = 10; o < 15; ++o) {
    row[o] = 0.f;
    col[o * 16] = 0.f;
  }
  row[15] = 1.0f;
  col[15 * 16] = 1.0f;
}

// ---------------------------------------------------------------------------
// Kernel 2: streaming (flash-style) attention. One wave owns a 16-query tile.
// Logits are bounded (|pooled| small) -> exp() directly, no running max.
// O (16x16 f32, D-layout) accumulates P*V; column 15 = softmax denominator.
// ---------------------------------------------------------------------------
__global__ __launch_bounds__(256) void attn_kernel(
    const float* __restrict__ pp, const float* __restrict__ ppT,
    float* __restrict__ out, int nTiles) {
  // Per-wave 16x16 transpose slab, row stride 18 dwords (bank-conflict free,
  // keeps 8-byte alignment for ds_load_b64).
  __shared__ float ldsbuf[8 * 16 * 18];
  const int lane = threadIdx.x & 31;
  const int wave = threadIdx.x >> 5;
  const int h = lane >> 4;   // half-wave: selects K/M-group per WMMA layout
  const int r = lane & 15;   // row (A) / column (B,D) index
  const int qt = blockIdx.x * 8 + wave;
  float* P_lds = ldsbuf + wave * (16 * 18);

  // Q tile in A-layout: lane holds row r, K = 4*kk + 2*h + {0,1}
  const float* qrow = pp + (qt * 16 + r) * 16 + 2 * h;
  v2f aQ0 = *(const v2f*)(qrow + 0);
  v2f aQ1 = *(const v2f*)(qrow + 4);
  v2f aQ2 = *(const v2f*)(qrow + 8);
  v2f aQ3 = *(const v2f*)(qrow + 12);

  v8f O = {0.f, 0.f, 0.f, 0.f, 0.f, 0.f, 0.f, 0.f};

  for (int kt = 0; kt < nTiles; ++kt) {
    // B operand of Q@K^T: B[k][n] = pooled[key n][k]; lane holds key n = r,
    // k = 4*kk + 2*h + {0,1}  -> same addressing as the Q rows.
    const float* krow = pp + (kt * 16 + r) * 16 + 2 * h;
    v2f bK0 = *(const v2f*)(krow + 0);
    v2f bK1 = *(const v2f*)(krow + 4);
    v2f bK2 = *(const v2f*)(krow + 8);
    v2f bK3 = *(const v2f*)(krow + 12);

    v8f S = {0.f, 0.f, 0.f, 0.f, 0.f, 0.f, 0.f, 0.f};
    S = wmma4(aQ0, bK0, S);
    S = wmma4(aQ1, bK1, S);
    S = wmma4(aQ2, bK2, S);
    S = wmma4(aQ3, bK3, S);

    // P = exp(S); write D-layout (row = j + 8h, col = r) to LDS
#pragma unroll
    for (int j = 0; j < 8; ++j)
      P_lds[(j + 8 * h) * 18 + r] = __expf(S[j]);
    asm volatile("s_wait_dscnt 0" ::: "memory");

    // Reload P in A-layout: lane holds row r, K = 4*kk + 2*h + {0,1}
    const float* prow = P_lds + r * 18 + 2 * h;
    v2f aP0 = *(const v2f*)(prow + 0);
    v2f aP1 = *(const v2f*)(prow + 4);
    v2f aP2 = *(const v2f*)(prow + 8);
    v2f aP3 = *(const v2f*)(prow + 12);

    // B operand of P@V: B[n][c] = pooled[key n][c]; lane holds dim c = r,
    // n = 4*kk + 2*h + {0,1} -> contiguous in the transposed copy ppT.
    const float* vrow = ppT + kt * 256 + r * 16 + 2 * h;
    v2f bV0 = *(const v2f*)(vrow + 0);
    v2f bV1 = *(const v2f*)(vrow + 4);
    v2f bV2 = *(const v2f*)(vrow + 8);
    v2f bV3 = *(const v2f*)(vrow + 12);

    O = wmma4(aP0, bV0, O);
    O = wmma4(aP1, bV1, O);
    O = wmma4(aP2, bV2, O);
    O = wmma4(aP3, bV3, O);
  }

  // Finalize: v[row][c] = O[row][c] / O[row][15]; only dims 0..9 are real.
#pragma unroll
  for (int j = 0; j < 8; ++j) {
    float oj = O[j];
    float denom = __shfl(oj, 15, 16);  // broadcast column 15 within 16-group
    if (r < 10) out[(qt * 16 + j + 8 * h) * 10 + r] = oj / denom;
  }
}

// ---------------------------------------------------------------------------
extern "C" void kernel_launch(void* const* d_in, const int* in_sizes, int n_in,
                              void* d_out, int out_size, void* d_ws,
                              size_t ws_size, hipStream_t stream) {
  const float* x = (const float*)d_in[0];  // [B,3,10]
  const float* A = (const float*)d_in[1];  // [3,3]
  const float* W = (const float*)d_in[2];  // [10,10]
  float* out = (float*)d_out;              // [B,10]
  int B = in_sizes[0] / 30;                // 8192

  float* pp = (float*)d_ws;                // [B][16] padded pooled
  float* ppT = pp + (size_t)B * 16;        // [B/16][16][16] tile-transposed

  pooled_kernel<<<(B + 255) / 256, 256, 0, stream>>>(x, A, W, pp, ppT, B);

  int nTiles = B / 16;                     // 512 key tiles
  attn_kernel<<<nTiles / 8, 256, 0, stream>>>(pp, ppT, out, nTiles);
}